// Addparam_25701084299720
// MI455X (gfx1250) — compile-verified
//
#include <hip/hip_runtime.h>
#include <hip/hip_bf16.h>

typedef __attribute__((ext_vector_type(2))) float v2f;
typedef __attribute__((ext_vector_type(8))) float v8f;

#define K_NEIGHBORS 8
#define COS_THRESH 0.8f
#define COS_EPS 1e-6f
#define LOG2E 1.44269504088896340736f

// ---------------------------------------------------------------------------
// Kernel 1: per-query top-8 nearest voxels -> mean normal (+ its length).
// One wave32 per query. Per-lane sorted top-8 over a strided scan, then 8
// rounds of lexicographic (d2, idx) wave-argmin via shuffles.
// ---------------------------------------------------------------------------
__global__ __launch_bounds__(256)
void knn_mean_normal_kernel(const float* __restrict__ xw,     // [N,3]
                            const float* __restrict__ vp,     // [V,6]
                            const float* __restrict__ vnrm,   // [V,3]
                            float* __restrict__ xn_out,       // [N,4] ws
                            int N, int V)
{
    const int lane = threadIdx.x & 31;
    const int q = (blockIdx.x * blockDim.x + threadIdx.x) >> 5;  // query id
    if (q >= N) return;  // whole wave exits together

    const float qx = xw[q * 3 + 0];
    const float qy = xw[q * 3 + 1];
    const float qz = xw[q * 3 + 2];

    float best[K_NEIGHBORS];
    int   bidx[K_NEIGHBORS];
#pragma unroll
    for (int k = 0; k < K_NEIGHBORS; ++k) { best[k] = 3.0e38f; bidx[k] = 0x7fffffff; }

    for (int j = lane; j < V; j += 32) {
        const float dx = qx - vp[j * 6 + 0];
        const float dy = qy - vp[j * 6 + 1];
        const float dz = qz - vp[j * 6 + 2];
        const float d2 = fmaf(dx, dx, fmaf(dy, dy, dz * dz));
        if (d2 < best[K_NEIGHBORS - 1]) {
            best[K_NEIGHBORS - 1] = d2;
            bidx[K_NEIGHBORS - 1] = j;
#pragma unroll
            for (int k = K_NEIGHBORS - 1; k > 0; --k) {
                if (best[k] < best[k - 1]) {
                    float tf = best[k]; best[k] = best[k - 1]; best[k - 1] = tf;
                    int   ti = bidx[k]; bidx[k] = bidx[k - 1]; bidx[k - 1] = ti;
                }
            }
        }
    }

    // Merge per-lane sorted lists: 8 rounds of wave argmin on (d2, idx).
    int hp = 0;
    float nx = 0.f, ny = 0.f, nz = 0.f;
#pragma unroll
    for (int r = 0; r < K_NEIGHBORS; ++r) {
        float m  = (hp < K_NEIGHBORS) ? best[hp] : 3.0e38f;
        int   mi = (hp < K_NEIGHBORS) ? bidx[hp] : 0x7fffffff;
#pragma unroll
        for (int off = 16; off >= 1; off >>= 1) {
            const float om = __shfl_xor(m, off, 32);
            const int   oi = __shfl_xor(mi, off, 32);
            if (om < m || (om == m && oi < mi)) { m = om; mi = oi; }
        }
        // all lanes agree on (m, mi): accumulate that voxel's normal
        nx += vnrm[mi * 3 + 0];
        ny += vnrm[mi * 3 + 1];
        nz += vnrm[mi * 3 + 2];
        if (hp < K_NEIGHBORS && bidx[hp] == mi) hp++;  // winning lane pops
    }

    if (lane == 0) {
        nx *= (1.0f / K_NEIGHBORS);
        ny *= (1.0f / K_NEIGHBORS);
        nz *= (1.0f / K_NEIGHBORS);
        const float len =
            __builtin_amdgcn_sqrtf(fmaf(nx, nx, fmaf(ny, ny, nz * nz)));
        xn_out[q * 4 + 0] = nx;
        xn_out[q * 4 + 1] = ny;
        xn_out[q * 4 + 2] = nz;
        xn_out[q * 4 + 3] = len;
    }
}

// ---------------------------------------------------------------------------
// Kernel 2: field pass using V_WMMA_F32_16X16X4_F32.
// Block = one 16-query row block; 8 waves split the V columns (16 per tile).
// WMMA #1: dist^2 = (x,1) . (-2p, p.p)  (+ per-row ||x||^2)
// WMMA #2: cosine numerator = x_normal . voxel_normal
// Epilogue uses raw v_sqrt_f32 / v_exp_f32 (one trans op each).
// ---------------------------------------------------------------------------
__global__ __launch_bounds__(256)
void field_wmma_kernel(const float* __restrict__ xw,      // [N,3]
                       const float* __restrict__ vp,      // [V,6]
                       const float* __restrict__ vnrm,    // [V,3]
                       const float* __restrict__ score,   // [V]
                       const float* __restrict__ d_a,     // [1]
                       const float* __restrict__ d_b,     // [1]
                       const float* __restrict__ xn,      // [N,4] ws
                       float* __restrict__ out,           // [2N]
                       int N, int V)
{
    __shared__ float s_sum[16];
    __shared__ float s_cnt[16];
    const int tid = threadIdx.x;
    if (tid < 16) { s_sum[tid] = 0.f; s_cnt[tid] = 0.f; }
    __syncthreads();

    const int lane = tid & 31;
    const int wv   = tid >> 5;           // wave 0..7 in block
    const int cl   = lane & 15;          // column within tile / A-row
    const int half = lane >> 4;          // K-half for A/B, row-half for C
    const int rowBase = blockIdx.x * 16;

    const float da = d_a[0];
    const float db = d_b[0];
    const float inv_da = 1.0f / da;
    const float ek = -db * LOG2E;        // exp(-db*d) == exp2(ek*d)

    // --- A matrices (16x4 f32 layout: lanes 0-15 hold K=0,1; 16-31 K=2,3) ---
    const int rowA = rowBase + cl;
    const float ax0 = xw[rowA * 3 + 0];
    const float ax1 = xw[rowA * 3 + 1];
    const float ax2 = xw[rowA * 3 + 2];
    const float an0 = xn[rowA * 4 + 0];
    const float an1 = xn[rowA * 4 + 1];
    const float an2 = xn[rowA * 4 + 2];
    v2f A, A2;
    A[0]  = half ? ax2 : ax0;
    A[1]  = half ? 1.0f : ax1;
    A2[0] = half ? an2 : an0;
    A2[1] = half ? 0.0f : an1;

    // Per-lane row-dependent scalars for the C rows this lane holds:
    // C VGPR r  <->  row (rowBase + 8*half + r)
    float xsq[8], xnl08[8];
#pragma unroll
    for (int r = 0; r < 8; ++r) {
        const int qr = rowBase + 8 * half + r;
        const float a0 = xw[qr * 3 + 0];
        const float a1 = xw[qr * 3 + 1];
        const float a2 = xw[qr * 3 + 2];
        xsq[r]   = fmaf(a0, a0, fmaf(a1, a1, a2 * a2));
        xnl08[r] = COS_THRESH * xn[qr * 4 + 3];      // 0.8*|x_normal|
    }
    const float thrEps = COS_THRESH * COS_EPS;        // 0.8*eps

    float ssum[8], scnt[8];
#pragma unroll
    for (int r = 0; r < 8; ++r) { ssum[r] = 0.f; scnt[r] = 0.f; }

    const int colsPerWave = V >> 3;               // 2048
    const int c0 = wv * colsPerWave;
    const int cEnd = c0 + colsPerWave;

    for (int cb = c0; cb < cEnd; cb += 16) {
        const int j = cb + cl;
        // prefetch the next tile's voxel row (speculative, dropped if OOB)
        __builtin_prefetch(vp + (size_t)(j + 16) * 6, 0, 0);

        // voxel point: 24B row -> aligned float2 + float
        const float2 p01 = *(const float2*)(vp + (size_t)j * 6);
        const float  p2  = vp[(size_t)j * 6 + 2];
        const float  n0  = vnrm[j * 3 + 0];
        const float  n1  = vnrm[j * 3 + 1];
        const float  n2  = vnrm[j * 3 + 2];
        const float  w_sc = score[j] * inv_da;        // hoisted out of r-loop

        const float pp  = fmaf(p01.x, p01.x, fmaf(p01.y, p01.y, p2 * p2));
        const float vnl =
            __builtin_amdgcn_sqrtf(fmaf(n0, n0, fmaf(n1, n1, n2 * n2)));

        v2f B, B2;
        B[0]  = half ? (-2.0f * p2) : (-2.0f * p01.x);
        B[1]  = half ? pp           : (-2.0f * p01.y);
        B2[0] = half ? n2   : n0;
        B2[1] = half ? 0.0f : n1;

        v8f zero = {};
        // D = A x B + 0  (distance^2 partial), 16x16x4 FP32 WMMA
        v8f Cd = __builtin_amdgcn_wmma_f32_16x16x4_f32(
            false, A, false, B, (short)0, zero, false, false);
        // D = A2 x B2 + 0 (cosine numerator)
        v8f Cn = __builtin_amdgcn_wmma_f32_16x16x4_f32(
            false, A2, false, B2, (short)0, zero, false, false);

#pragma unroll
        for (int r = 0; r < 8; ++r) {
            const float d2  = Cd[r] + xsq[r];
            const float d   = __builtin_amdgcn_sqrtf(fmaxf(d2, 0.0f));
            const float thr = fmaxf(xnl08[r] * vnl, thrEps);
            const bool  msk = Cn[r] > thr;            // cos > 0.8
            const float w   = msk ? (w_sc * __builtin_amdgcn_exp2f(ek * d))
                                  : 0.0f;
            ssum[r] += w;
            scnt[r] += msk ? 1.0f : 0.0f;
        }
    }

    // Reduce across the 16 lanes that share each row half.
#pragma unroll
    for (int off = 1; off <= 8; off <<= 1) {
#pragma unroll
        for (int r = 0; r < 8; ++r) {
            ssum[r] += __shfl_xor(ssum[r], off, 32);
            scnt[r] += __shfl_xor(scnt[r], off, 32);
        }
    }

    if (cl == 0) {  // lanes 0 (rows 0..7) and 16 (rows 8..15)
#pragma unroll
        for (int r = 0; r < 8; ++r) {
            atomicAdd(&s_sum[8 * half + r], ssum[r]);   // ds_add_f32
            atomicAdd(&s_cnt[8 * half + r], scnt[r]);
        }
    }
    __syncthreads();

    if (tid < 16) {
        const int q = rowBase + tid;
        const float c = s_cnt[tid];
        const bool nz = c > 0.5f;
        out[q]     = nz ? (s_sum[tid] / c) : 0.0f;   // score_sum / score_num
        out[N + q] = nz ? 1.0f : 0.0f;               // nonzero mask
    }
}

// ---------------------------------------------------------------------------
extern "C" void kernel_launch(void* const* d_in, const int* in_sizes, int n_in,
                              void* d_out, int out_size, void* d_ws, size_t ws_size,
                              hipStream_t stream) {
    const float* xw   = (const float*)d_in[0];  // x_world      [N,1,3]
    const float* vp   = (const float*)d_in[1];  // voxel_point  [1,V,6]
    const float* vnrm = (const float*)d_in[2];  // voxel_normal [V,3]
    const float* sc   = (const float*)d_in[3];  // score        [V]
    const float* da   = (const float*)d_in[4];  // d_a          [1]
    const float* db   = (const float*)d_in[5];  // d_b          [1]

    const int N = in_sizes[0] / 3;   // 4096
    const int V = in_sizes[3];       // 16384

    float* xn = (float*)d_ws;        // [N,4] mean normal + length
    float* out = (float*)d_out;      // [2N]: field then nonzero(0/1)

    // Kernel 1: 8 waves (queries) per 256-thread block.
    knn_mean_normal_kernel<<<(N + 7) / 8, 256, 0, stream>>>(xw, vp, vnrm, xn, N, V);

    // Kernel 2: one block per 16-query row block; 8 waves split V columns.
    field_wmma_kernel<<<N / 16, 256, 0, stream>>>(xw, vp, vnrm, sc, da, db, xn,
                                                  out, N, V);
}